// EfficientFourierConvSSM_13091060318291
// MI455X (gfx1250) — compile-verified
//
#include <hip/hip_runtime.h>
#include <math.h>

// ---------------------------------------------------------------------------
// EfficientFourierConvSSM, collapsed: out = S * B_f * x,  S = sum_{k<16} A_f^k
//   = (1+A)(1+A^2)(1+A^4)(1+A^8)    (complex, elementwise per (h,w,c))
// Memory-bound streaming pass (285 MB @ 23.3 TB/s ~ 12 us); DFT of the 7x7
// kernels is done with V_WMMA_F32_16X16X4_F32; multiplier row staged to LDS
// with the Tensor Data Mover.
// ---------------------------------------------------------------------------

#define Hdim 128
#define Wdim 128
#define Cdim 128
#define Bdim 8
#define Kk   7
#define GST  8            // K padded 7 -> 8 for two chained K=4 WMMAs

typedef __attribute__((ext_vector_type(2))) float        v2f;
typedef __attribute__((ext_vector_type(4))) float        f4;
typedef __attribute__((ext_vector_type(8))) float        v8f;
typedef __attribute__((ext_vector_type(4))) unsigned int v4u;
typedef __attribute__((ext_vector_type(4))) int          v4i;
typedef __attribute__((ext_vector_type(8))) int          v8i;

#define TWO_PI_F 6.28318530717958647692f

#if defined(__has_builtin)
#if __has_builtin(__builtin_amdgcn_wmma_f32_16x16x4_f32)
#define HAVE_WMMA_F32 1
#endif
#if __has_builtin(__builtin_amdgcn_tensor_load_to_lds) && __has_builtin(__builtin_amdgcn_s_wait_tensorcnt)
#define HAVE_TDM 1
#endif
#endif

// ---------------------------------------------------------------------------
// Stage 1: row DFT of the two 7x7 kernels.
//   G[c,u,kx] = sum_ky kern[c,ky,kx] * exp(-2*pi*i * u*(ky-3)/128)
// A gets the decay mask + 0.9*tanh first. Output stride-8, tap 7 zeroed so the
// WMMA K=8 contraction is exact. 16384 threads, trivial cost.
// ---------------------------------------------------------------------------
__global__ __launch_bounds__(256)
void stage1_rowdft(const float* __restrict__ A, const float* __restrict__ Bk,
                   float* __restrict__ GAr, float* __restrict__ GAi,
                   float* __restrict__ GBr, float* __restrict__ GBi) {
    int idx = blockIdx.x * blockDim.x + threadIdx.x;     // idx = c*128 + u
    if (idx >= Cdim * Hdim) return;
    int c = idx >> 7;
    int u = idx & 127;
    for (int kx = 0; kx < Kk; ++kx) {
        float sra = 0.f, sia = 0.f, srb = 0.f, sib = 0.f;
        for (int ky = 0; ky < Kk; ++ky) {
            float dy = (float)(ky - 3), dx = (float)(kx - 3);
            float decay = __expf(-0.3f * sqrtf(dy * dy + dx * dx) / 3.0f);
            float a = 0.9f * tanhf(A[(c * 7 + ky) * 7 + kx] * decay);
            float b = Bk[(c * 7 + ky) * 7 + kx];
            float ang = -TWO_PI_F * (float)(u * (ky - 3)) * (1.0f / 128.0f);
            float sn, cs;
            __sincosf(ang, &sn, &cs);
            sra += a * cs; sia += a * sn;
            srb += b * cs; sib += b * sn;
        }
        GAr[idx * GST + kx] = sra; GAi[idx * GST + kx] = sia;
        GBr[idx * GST + kx] = srb; GBi[idx * GST + kx] = sib;
    }
    GAr[idx * GST + 7] = 0.f; GAi[idx * GST + 7] = 0.f;   // K-pad
    GBr[idx * GST + 7] = 0.f; GBi[idx * GST + 7] = 0.f;
}

#ifdef HAVE_WMMA_F32
// A-operand layout (32-bit A 16x4): lanes 0-15 row M, v0={K0|K2}, v1={K1|K3}.
static __device__ __forceinline__ v2f ld2(const float* __restrict__ g, int base, int k0) {
    v2f r; r.x = g[base + k0]; r.y = g[base + k0 + 1]; return r;
}
#endif

// ---------------------------------------------------------------------------
// Stage 2: column DFT as a matmul  (16384 x 8) . (8 x 128)  per {A,B}x{r,i},
// one wave per 16x16 tile of (c,u) x v, V_WMMA_F32_16X16X4_F32 chained K=4+4.
// Then per-lane complex geometric series S and fused M = S*B_f, stored in the
// x-compatible (H,W,2C) layout.
// ---------------------------------------------------------------------------
__global__ __launch_bounds__(256)
void stage2_multiplier(const float* __restrict__ GAr, const float* __restrict__ GAi,
                       const float* __restrict__ GBr, const float* __restrict__ GBi,
                       float* __restrict__ Mout) {
    int wid   = (blockIdx.x * blockDim.x + threadIdx.x) >> 5;  // 8192 tiles
    int lane  = threadIdx.x & 31;
    int tileR = wid >> 3;            // 1024 row-tiles over (c,u)
    int tileV = wid & 7;             // 8 col-tiles over v
    int R0    = tileR << 4;
    int v0    = tileV << 4;
    int c     = R0 >> 7;             // 128 rows per channel; tiles never straddle c
    int u0    = R0 & 127;
    int n     = lane & 15;
    int khalf = lane >> 4;
    int v     = v0 + n;

    // Twiddles for this lane's K slots: W[k][v] = exp(-2*pi*i * v*(k-3)/128)
    float wr[4], wi[4];
    {
        int ks[4] = {2 * khalf, 2 * khalf + 1, 4 + 2 * khalf, 5 + 2 * khalf};
        for (int t = 0; t < 4; ++t) {
            float ang = -TWO_PI_F * (float)(v * (ks[t] - 3)) * (1.0f / 128.0f);
            __sincosf(ang, &wi[t], &wr[t]);
        }
    }

    v8f afr = {}, afi = {}, bfr = {}, bfi = {};

#ifdef HAVE_WMMA_F32
    int rbase = (R0 + n) * GST;
    v2f arlo = ld2(GAr, rbase, 2 * khalf), arhi = ld2(GAr, rbase, 4 + 2 * khalf);
    v2f ailo = ld2(GAi, rbase, 2 * khalf), aihi = ld2(GAi, rbase, 4 + 2 * khalf);
    v2f brlo = ld2(GBr, rbase, 2 * khalf), brhi = ld2(GBr, rbase, 4 + 2 * khalf);
    v2f bilo = ld2(GBi, rbase, 2 * khalf), bihi = ld2(GBi, rbase, 4 + 2 * khalf);
    v2f wrlo = {wr[0], wr[1]}, wrhi = {wr[2], wr[3]};
    v2f wilo = {wi[0], wi[1]}, wihi = {wi[2], wi[3]};
    v2f nwlo = {-wi[0], -wi[1]}, nwhi = {-wi[2], -wi[3]};

#define WMMA4(ACC, Aop, Bop) \
    ACC = __builtin_amdgcn_wmma_f32_16x16x4_f32(false, Aop, false, Bop, (short)0, ACC, false, false)

    // A_f_r = Gr.Wr - Gi.Wi   A_f_i = Gr.Wi + Gi.Wr   (same for B kernel)
    WMMA4(afr, arlo, wrlo); WMMA4(afr, arhi, wrhi); WMMA4(afr, ailo, nwlo); WMMA4(afr, aihi, nwhi);
    WMMA4(afi, arlo, wilo); WMMA4(afi, arhi, wihi); WMMA4(afi, ailo, wrlo); WMMA4(afi, aihi, wrhi);
    WMMA4(bfr, brlo, wrlo); WMMA4(bfr, brhi, wrhi); WMMA4(bfr, bilo, nwlo); WMMA4(bfr, bihi, nwhi);
    WMMA4(bfi, brlo, wilo); WMMA4(bfi, brhi, wihi); WMMA4(bfi, bilo, wrlo); WMMA4(bfi, bihi, wrhi);
#undef WMMA4
#else
    // Scalar fallback: direct 7-tap column DFT per output element.
    for (int j = 0; j < 8; ++j) {
        int R = R0 + j + 8 * khalf;
        float ar = 0.f, ai = 0.f, br = 0.f, bi = 0.f;
        for (int k = 0; k < Kk; ++k) {
            float ang = -TWO_PI_F * (float)(v * (k - 3)) * (1.0f / 128.0f);
            float sn, cs; __sincosf(ang, &sn, &cs);
            float gr = GAr[R * GST + k], gi = GAi[R * GST + k];
            ar += gr * cs - gi * sn;  ai += gr * sn + gi * cs;
            float hr = GBr[R * GST + k], hi = GBi[R * GST + k];
            br += hr * cs - hi * sn;  bi += hr * sn + hi * cs;
        }
        afr[j] = ar; afi[j] = ai; bfr[j] = br; bfi[j] = bi;
    }
#endif

    // C/D layout: reg j -> row M = j + 8*(lane>=16), col N = lane&15.
    for (int j = 0; j < 8; ++j) {
        float ar = afr[j], ai = afi[j];
        float br = bfr[j], bi = bfi[j];
        // S = (1+A)(1+A^2)(1+A^4)(1+A^8) : geometric sum of 16 terms
        float sr = 1.f + ar, si = ai;
        float pr = ar, pi = ai;
        for (int t = 0; t < 3; ++t) {
            float qr = pr * pr - pi * pi;
            float qi = 2.f * pr * pi;
            pr = qr; pi = qi;                       // A^(2^(t+1))
            float tr = 1.f + pr, ti = pi;
            float xr = sr * tr - si * ti;
            float xi = sr * ti + si * tr;
            sr = xr; si = xi;
        }
        float mr = sr * br - si * bi;               // M = S * B_f
        float mi = sr * bi + si * br;
        int urow = u0 + j + 8 * khalf;
        int base = ((urow << 7) + v) << 8;          // (u*128+v)*256
        Mout[base + c]       = mr;
        Mout[base + 128 + c] = mi;
    }
}

// ---------------------------------------------------------------------------
// Stage 3: the bandwidth-bound pass.  One block per (u,v); the 1 KB multiplier
// row is staged into LDS by the Tensor Data Mover (reused by all 8 batches),
// then 8 waves stream x -> out with nontemporal complex multiplies.
// ---------------------------------------------------------------------------
__global__ __launch_bounds__(256)
void fused_apply(const float* __restrict__ x, const float* __restrict__ Mmul,
                 float* __restrict__ out) {
    __shared__ float smem[256];
    const int uv  = blockIdx.x;          // 0 .. 16383
    const int tid = threadIdx.x;
    const float* __restrict__ Mrow = Mmul + (size_t)uv * 256;

#ifdef HAVE_TDM
    if (tid < 32) {                       // one wave builds D# and issues the TDM
        unsigned int       lds_off = (unsigned int)(unsigned long long)(uintptr_t)smem;
        unsigned long long ga      = (unsigned long long)(uintptr_t)Mrow;
        v4u g0;
        g0.x = 1u;                                                // count=1, user D#
        g0.y = lds_off;                                           // lds_addr (bytes)
        g0.z = (unsigned int)(ga & 0xFFFFFFFFull);                // global_addr[31:0]
        g0.w = (unsigned int)((ga >> 32) & 0x1FFFFFFull) | (2u << 30); // [56:32] | type=2
        v8i g1;
        g1[0] = (int)(2u << 16);      // data_size=4B, workgroup_mask=0
        g1[1] = (int)(256u << 16);    // tensor_dim0 = 256 elements
        g1[2] = (int)(1u << 16);      // tensor_dim1 = 1
        g1[3] = (int)(256u << 16);    // tile_dim0 = 256
        g1[4] = 1;                    // tile_dim1 = 1, tile_dim2 = 0
        g1[5] = 256;                  // tensor_dim0_stride = 256
        g1[6] = 0;
        g1[7] = 0;
        v4i z4 = {0, 0, 0, 0};
        v8i z8 = {0, 0, 0, 0, 0, 0, 0, 0};
        // clang-23 / therock-10.0 arity: (g0, g1, g2, g3, g4, cpol)
        __builtin_amdgcn_tensor_load_to_lds(g0, g1, z4, z4, z8, 0);
        __builtin_amdgcn_s_wait_tensorcnt(0);
    }
    __syncthreads();
#else
    if (tid < 64) ((f4*)smem)[tid] = ((const f4*)Mrow)[tid];
    __syncthreads();
#endif

    // tid -> (batch, float4-channel); one wave covers one contiguous 512B half-row
    const int b  = tid >> 5;             // 0..7
    const int c4 = tid & 31;             // 0..31 float4s of 128 channels
    const size_t rowbase = (size_t)(b * (Hdim * Wdim) + uv) * 256;

    const f4* __restrict__ xin = (const f4*)(x + rowbase);
    f4 xr = __builtin_nontemporal_load(&xin[c4]);
    f4 xi = __builtin_nontemporal_load(&xin[32 + c4]);
    f4 mr = ((const f4*)smem)[c4];
    f4 mi = ((const f4*)smem)[32 + c4];

    f4 o_r = mr * xr - mi * xi;
    f4 o_i = mr * xi + mi * xr;

    f4* __restrict__ op = (f4*)(out + rowbase);
    __builtin_nontemporal_store(o_r, &op[c4]);
    __builtin_nontemporal_store(o_i, &op[32 + c4]);
}

// ---------------------------------------------------------------------------
extern "C" void kernel_launch(void* const* d_in, const int* in_sizes, int n_in,
                              void* d_out, int out_size, void* d_ws, size_t ws_size,
                              hipStream_t stream) {
    const float* x  = (const float*)d_in[0];   // (8,128,128,256) f32
    const float* A  = (const float*)d_in[1];   // (128,7,7) f32
    const float* Bk = (const float*)d_in[2];   // (128,7,7) f32
    float* out = (float*)d_out;
    float* ws  = (float*)d_ws;

    float* GAr  = ws;                 // 16384*8 floats each
    float* GAi  = ws + 131072;
    float* GBr  = ws + 262144;
    float* GBi  = ws + 393216;
    float* Mmul = ws + 524288;        // 128*128*256 floats = 16 MB

    stage1_rowdft<<<64, 256, 0, stream>>>(A, Bk, GAr, GAi, GBr, GBi);
    stage2_multiplier<<<1024, 256, 0, stream>>>(GAr, GAi, GBr, GBi, Mmul);
    fused_apply<<<Hdim * Wdim, 256, 0, stream>>>(x, Mmul, out);
}